// GATNet_11063835754883
// MI455X (gfx1250) — compile-verified
//
#include <hip/hip_runtime.h>
#include <cstdint>

#define NN   100000
#define EE   1600000
#define ETOT (EE + NN)
#define FIN  128
#define F1   64      // H1*C1
#define H1N  8
#define C1N  8
#define F2   16      // H2*C2
#define NEG  0.2f
#define SENT 0x007FFFFFu   // ordered-int encoding of -inf

#define XS_STRIDE  (FIN + 4)   // 132: TDM-padded A-tile stride, conflict-free b64
#define WT_STRIDE  (FIN + 4)   // 132: transposed W1 stride
#define WT2_STRIDE (F1 + 4)    // 68 : transposed W2 stride

typedef float    v2f __attribute__((ext_vector_type(2)));
typedef float    v8f __attribute__((ext_vector_type(8)));
typedef unsigned v4u __attribute__((ext_vector_type(4)));
typedef int      v4i __attribute__((ext_vector_type(4)));
typedef int      v8i __attribute__((ext_vector_type(8)));

// monotone float <-> uint mapping so unsigned atomicMax == float max
__device__ __forceinline__ unsigned fenc(float f) {
    int i = __float_as_int(f);
    return (unsigned)(i >= 0 ? (i ^ 0x80000000) : ~i);
}
__device__ __forceinline__ float fdec(unsigned u) {
    int i = (u & 0x80000000u) ? (int)(u ^ 0x80000000u) : (int)(~u);
    return __int_as_float(i);
}
__device__ __forceinline__ float lrelu(float x) { return x >= 0.f ? x : NEG * x; }

// ---- Tensor Data Mover: DMA a rows x width fp32 tile (row stride `stride` elts)
// from global memory into LDS, optionally inserting pad dwords per row so the
// LDS image has a bank-conflict-free stride. D# built per CDNA5 ISA ch.8.
__device__ __forceinline__ void tdm_load_2d(unsigned lds_bytes, const void* gsrc,
                                            unsigned width, unsigned rows, unsigned stride,
                                            unsigned pad_interval, unsigned pad_amount,
                                            bool pad_en) {
    unsigned long long ga = (unsigned long long)gsrc;
    v4u g0 = { 1u,                                   // count=1 (valid), user mode
               lds_bytes,                            // lds_addr
               (unsigned)ga,                         // global_addr[31:0]
               (unsigned)((ga >> 32) & 0x01FFFFFFu) | (2u << 30) };  // addr[56:32], type=2
    v8i g1;
    g1[0] = (int)((2u << 16) |                       // data_size = 4B
                  (pad_en ? (1u << 20) : 0u) |
                  (pad_interval << 22) | (pad_amount << 25));
    g1[1] = (int)(width << 16);                      // tensor_dim0[15:0]
    g1[2] = (int)((width >> 16) | (rows << 16));     // tensor_dim0[31:16] | tensor_dim1[15:0]
    g1[3] = (int)((rows  >> 16) | (width << 16));    // tensor_dim1[31:16] | tile_dim0
    g1[4] = (int)rows;                               // tile_dim1 (tile_dim2 = 0)
    g1[5] = (int)stride;                             // tensor_dim0_stride[31:0]
    g1[6] = 0;
    g1[7] = 0;
    v4i z4 = {0, 0, 0, 0};
#if defined(__clang_major__) && __clang_major__ >= 23
    v8i z8 = {0, 0, 0, 0, 0, 0, 0, 0};
    __builtin_amdgcn_tensor_load_to_lds(g0, g1, z4, z4, z8, 0);
#else
    __builtin_amdgcn_tensor_load_to_lds(g0, g1, z4, z4, 0);
#endif
}

// ---------------- init workspace / output ----------------
__global__ __launch_bounds__(256) void k_init(float* denom1, unsigned* emax1, float* acc1,
                                              float* denom2, unsigned* emax2, float* out) {
    int i = blockIdx.x * blockDim.x + threadIdx.x;
    if (i < NN * F1) acc1[i] = 0.f;
    if (i < NN * H1N) { denom1[i] = 0.f; emax1[i] = SENT; }
    if (i < NN * F2) out[i] = 0.f;
    if (i < NN) { denom2[i] = 0.f; emax2[i] = SENT; }
}

// ---------------- GEMM1: h1[N,64] = x[N,128] @ W1[128,64], fp32 WMMA ----------------
// 1 block = 16 rows; A tile DMA'd by the TDM with row padding (stride 132 -> no bank
// conflicts), W1 staged transposed so B K-pairs are contiguous ds_load_b64s.
__global__ __launch_bounds__(128) void k_gemm1(const float* __restrict__ x,
                                               const float* __restrict__ W,
                                               float* __restrict__ h1) {
    __shared__ float xs[16 * XS_STRIDE];      // TDM-padded 16x128 A tile
    __shared__ float wt[F1 * WT_STRIDE];      // W1^T : wt[c*132 + k]
    const int row0 = blockIdx.x * 16;
    const int tid  = threadIdx.x;

    if (tid < 32) {   // wave 0 issues one tensor DMA; pad 4 dwords every 128 dwords
        tdm_load_2d((unsigned)(unsigned long long)xs, x + (size_t)row0 * FIN,
                    FIN, 16, FIN, /*pad_interval=128dw*/6, /*pad_amount=4dw*/3, true);
        __builtin_amdgcn_s_wait_tensorcnt(0);
    }
    for (int i = tid; i < FIN * F1; i += 128) {          // coalesced read of W
        int r = i >> 6, c = i & 63;                      // W[r][c]
        wt[c * WT_STRIDE + r] = W[i];
    }
    __syncthreads();

    const int lane    = tid & 31;
    const int colBase = (tid >> 5) * 16;
    const int m       = lane & 15;                       // A row / B col within tile
    const int kh      = (lane >> 4) * 2;                 // K sub-offset per lane half
    const float* brow = wt + (size_t)(colBase + m) * WT_STRIDE;

    v8f c = {};
    for (int k = 0; k < FIN; k += 4) {
        v2f a, b;
        a.x = xs[m * XS_STRIDE + k + kh];
        a.y = xs[m * XS_STRIDE + k + kh + 1];
        b.x = brow[k + kh];
        b.y = brow[k + kh + 1];
        c = __builtin_amdgcn_wmma_f32_16x16x4_f32(false, a, false, b, (short)0, c, false, false);
    }
    const int nCol  = colBase + m;
    const int rowHi = (lane >> 4) * 8;
#pragma unroll
    for (int r = 0; r < 8; ++r)
        h1[(size_t)(row0 + rowHi + r) * F1 + nCol] = c[r];
}

// ---------------- per-node attention logits, layer 1 ----------------
__global__ __launch_bounds__(256) void k_alpha1(const float* __restrict__ h1,
                                                const float* __restrict__ a_src,
                                                const float* __restrict__ a_dst,
                                                float* __restrict__ as, float* __restrict__ ad) {
    int i = blockIdx.x * blockDim.x + threadIdx.x;
    if (i >= NN * H1N) return;
    int n = i >> 3, h = i & 7;
    const float* row = h1 + (size_t)n * F1 + h * C1N;
    float s = 0.f, d = 0.f;
#pragma unroll
    for (int c = 0; c < C1N; ++c) {
        s += row[c] * a_src[h * C1N + c];
        d += row[c] * a_dst[h * C1N + c];
    }
    as[i] = s; ad[i] = d;
}

// ---------------- edge segment-max, layer 1 (thread per edge*head) ----------------
__global__ __launch_bounds__(256) void k_edgemax1(const long long* __restrict__ ei,
                                                  const float* __restrict__ as,
                                                  const float* __restrict__ ad,
                                                  unsigned* __restrict__ emax) {
    int i = blockIdx.x * blockDim.x + threadIdx.x;
    if (i >= ETOT * H1N) return;
    int t = i >> 3, h = i & 7;
    int s, d;
    if (t < EE) { s = (int)ei[t]; d = (int)ei[EE + t]; } else { s = d = t - EE; }
    float e = lrelu(as[s * H1N + h] + ad[d * H1N + h]);
    atomicMax(&emax[d * H1N + h], fenc(e));
}

// ---------------- edge accumulate (denom + messages), layer 1 ----------------
__global__ __launch_bounds__(256) void k_edgeacc1(const long long* __restrict__ ei,
                                                  const float* __restrict__ as,
                                                  const float* __restrict__ ad,
                                                  const unsigned* __restrict__ emax,
                                                  const float* __restrict__ h1,
                                                  float* __restrict__ denom,
                                                  float* __restrict__ acc) {
    int i = blockIdx.x * blockDim.x + threadIdx.x;
    if (i >= ETOT * H1N) return;
    int t = i >> 3, h = i & 7;
    int s, d;
    if (t < EE) { s = (int)ei[t]; d = (int)ei[EE + t]; } else { s = d = t - EE; }
    const float* hrow = h1 + (size_t)s * F1 + h * C1N;
    __builtin_prefetch(hrow, 0, 3);                 // global_prefetch: hide gather latency
    float e = lrelu(as[s * H1N + h] + ad[d * H1N + h]);
    float w = expf(e - fdec(emax[d * H1N + h]));
    atomicAdd(&denom[d * H1N + h], w);
    float* arow = acc + (size_t)d * F1 + h * C1N;
#pragma unroll
    for (int c = 0; c < C1N; ++c)
        atomicAdd(&arow[c], w * hrow[c]);           // contiguous 32B per edge-head
}

// ---------------- finalize layer 1: divide, bias, ELU ----------------
__global__ __launch_bounds__(256) void k_fin1(const float* __restrict__ acc,
                                              const float* __restrict__ denom,
                                              const float* __restrict__ b1,
                                              float* __restrict__ h2) {
    int i = blockIdx.x * blockDim.x + threadIdx.x;
    if (i >= NN * F1) return;
    int n = i >> 6, j = i & 63;
    float v = acc[i] / denom[n * H1N + (j >> 3)] + b1[j];
    h2[i] = v > 0.f ? v : expm1f(v);
}

// ---------------- GEMM2: g[N,16] = h2[N,64] @ W2[64,16], fp32 WMMA ----------------
// W2 staged transposed in LDS; 1 wave per 16-row block, 4 waves per block.
__global__ __launch_bounds__(128) void k_gemm2(const float* __restrict__ h2,
                                               const float* __restrict__ W2,
                                               float* __restrict__ g) {
    __shared__ float wt2[F2 * WT2_STRIDE];           // W2^T : wt2[c*68 + k]
    const int tid = threadIdx.x;
    for (int i = tid; i < F1 * F2; i += 128) {
        int r = i >> 4, c = i & 15;                  // W2[r][c]
        wt2[c * WT2_STRIDE + r] = W2[i];
    }
    __syncthreads();

    const int lane = tid & 31;
    const int rb   = blockIdx.x * 4 + (tid >> 5);
    if (rb >= NN / 16) return;                       // wave-uniform, after the only barrier
    const int m  = lane & 15;
    const int kh = (lane >> 4) * 2;
    const float* arow = h2 + (size_t)(rb * 16 + m) * F1;
    const float* brow = wt2 + (size_t)m * WT2_STRIDE;

    v8f c = {};
    for (int k = 0; k < F1; k += 4) {
        v2f a, b;
        a.x = arow[k + kh];
        a.y = arow[k + kh + 1];
        b.x = brow[k + kh];
        b.y = brow[k + kh + 1];
        c = __builtin_amdgcn_wmma_f32_16x16x4_f32(false, a, false, b, (short)0, c, false, false);
    }
    const int rowHi = (lane >> 4) * 8;
#pragma unroll
    for (int r = 0; r < 8; ++r)
        g[(size_t)(rb * 16 + rowHi + r) * F2 + m] = c[r];
}

// ---------------- per-node attention logits, layer 2 (1 head) ----------------
__global__ __launch_bounds__(256) void k_alpha2(const float* __restrict__ g,
                                                const float* __restrict__ a_src,
                                                const float* __restrict__ a_dst,
                                                float* __restrict__ as, float* __restrict__ ad) {
    int n = blockIdx.x * blockDim.x + threadIdx.x;
    if (n >= NN) return;
    float s = 0.f, d = 0.f;
#pragma unroll
    for (int c = 0; c < F2; ++c) {
        float v = g[(size_t)n * F2 + c];
        s += v * a_src[c];
        d += v * a_dst[c];
    }
    as[n] = s; ad[n] = d;
}

__global__ __launch_bounds__(256) void k_edgemax2(const long long* __restrict__ ei,
                                                  const float* __restrict__ as,
                                                  const float* __restrict__ ad,
                                                  unsigned* __restrict__ emax) {
    int t = blockIdx.x * blockDim.x + threadIdx.x;
    if (t >= ETOT) return;
    int s, d;
    if (t < EE) { s = (int)ei[t]; d = (int)ei[EE + t]; } else { s = d = t - EE; }
    float e = lrelu(as[s] + ad[d]);
    atomicMax(&emax[d], fenc(e));
}

__global__ __launch_bounds__(256) void k_edgeacc2(const long long* __restrict__ ei,
                                                  const float* __restrict__ as,
                                                  const float* __restrict__ ad,
                                                  const unsigned* __restrict__ emax,
                                                  const float* __restrict__ g,
                                                  float* __restrict__ denom,
                                                  float* __restrict__ out) {
    int t = blockIdx.x * blockDim.x + threadIdx.x;
    if (t >= ETOT) return;
    int s, d;
    if (t < EE) { s = (int)ei[t]; d = (int)ei[EE + t]; } else { s = d = t - EE; }
    const float* grow = g + (size_t)s * F2;
    __builtin_prefetch(grow, 0, 3);
    float e = lrelu(as[s] + ad[d]);
    float w = expf(e - fdec(emax[d]));
    atomicAdd(&denom[d], w);
    float* orow = out + (size_t)d * F2;
#pragma unroll
    for (int c = 0; c < F2; ++c)
        atomicAdd(&orow[c], w * grow[c]);
}

__global__ __launch_bounds__(256) void k_fin2(const float* __restrict__ denom,
                                              const float* __restrict__ b2,
                                              float* __restrict__ out) {
    int i = blockIdx.x * blockDim.x + threadIdx.x;
    if (i >= NN * F2) return;
    out[i] = out[i] / denom[i >> 4] + b2[i & 15];
}

static inline int cdiv(long long a, long long b) { return (int)((a + b - 1) / b); }

extern "C" void kernel_launch(void* const* d_in, const int* in_sizes, int n_in,
                              void* d_out, int out_size, void* d_ws, size_t ws_size,
                              hipStream_t stream) {
    const float*     x     = (const float*)d_in[0];
    const long long* ei    = (const long long*)d_in[1];
    const float*     W1    = (const float*)d_in[2];
    const float*     asrc1 = (const float*)d_in[3];
    const float*     adst1 = (const float*)d_in[4];
    const float*     b1    = (const float*)d_in[5];
    const float*     W2    = (const float*)d_in[6];
    const float*     asrc2 = (const float*)d_in[7];
    const float*     adst2 = (const float*)d_in[8];
    const float*     b2    = (const float*)d_in[9];
    float* out = (float*)d_out;

    float* ws = (float*)d_ws;
    size_t o = 0;
    float*    h1     = ws + o; o += (size_t)NN * F1;
    float*    as1    = ws + o; o += (size_t)NN * H1N;
    float*    ad1    = ws + o; o += (size_t)NN * H1N;
    unsigned* emax1  = (unsigned*)(ws + o); o += (size_t)NN * H1N;
    float*    denom1 = ws + o; o += (size_t)NN * H1N;
    float*    acc1   = ws + o; o += (size_t)NN * F1;
    float*    h2     = ws + o; o += (size_t)NN * F1;
    float*    g2     = ws + o; o += (size_t)NN * F2;
    float*    as2    = ws + o; o += (size_t)NN;
    float*    ad2    = ws + o; o += (size_t)NN;
    unsigned* emax2  = (unsigned*)(ws + o); o += (size_t)NN;
    float*    denom2 = ws + o; o += (size_t)NN;
    (void)ws_size; (void)in_sizes; (void)n_in; (void)out_size;

    k_init<<<cdiv((long long)NN * F1, 256), 256, 0, stream>>>(denom1, emax1, acc1, denom2, emax2, out);

    // layer 1
    k_gemm1<<<NN / 16, 128, 0, stream>>>(x, W1, h1);
    k_alpha1<<<cdiv((long long)NN * H1N, 256), 256, 0, stream>>>(h1, asrc1, adst1, as1, ad1);
    k_edgemax1<<<cdiv((long long)ETOT * H1N, 256), 256, 0, stream>>>(ei, as1, ad1, emax1);
    k_edgeacc1<<<cdiv((long long)ETOT * H1N, 256), 256, 0, stream>>>(ei, as1, ad1, emax1, h1, denom1, acc1);
    k_fin1<<<cdiv((long long)NN * F1, 256), 256, 0, stream>>>(acc1, denom1, b1, h2);

    // layer 2
    k_gemm2<<<cdiv(NN / 16, 4), 128, 0, stream>>>(h2, W2, g2);
    k_alpha2<<<cdiv(NN, 256), 256, 0, stream>>>(g2, asrc2, adst2, as2, ad2);
    k_edgemax2<<<cdiv(ETOT, 256), 256, 0, stream>>>(ei, as2, ad2, emax2);
    k_edgeacc2<<<cdiv(ETOT, 256), 256, 0, stream>>>(ei, as2, ad2, emax2, g2, denom2, out);
    k_fin2<<<cdiv((long long)NN * F2, 256), 256, 0, stream>>>(denom2, b2, out);
}